// SimSiamLoss_22084721836474
// MI455X (gfx1250) — compile-verified
//
#include <hip/hip_runtime.h>
#include <hip/hip_bf16.h>
#include <math.h>

// ---------------------------------------------------------------------------
// SimSiam subset loss, algebraically reduced:
//   pair_sum_s = sum_c S_p(c)·S_z(c) - sum_i pn_i·zn_i
//   cnt_s      = sum_c n_c(n_c-1)/2
//   loss       = mean_s( -0.5 * pair_sum_s / cnt_s )
// B=12288 rows (3 subsets x 4096), D=256, 64 classes.
// Memory-bound: 25 MB reads -> ~1.1us at 23.3 TB/s.
// ---------------------------------------------------------------------------

typedef __attribute__((ext_vector_type(2))) float v2f;
typedef __attribute__((ext_vector_type(8))) float v8f;

#define NSUB 3
#define MROWS 4096
#define DIMS 256
#define NCLS 64
#define CHUNKS 16              // blocks per subset
#define ROWS_PER_BLOCK (MROWS / CHUNKS)  // 256
#define SUMS (NCLS * DIMS)     // 16384 floats per array per subset

__device__ __forceinline__ float wred32(float v) {
  v += __shfl_xor(v, 1);
  v += __shfl_xor(v, 2);
  v += __shfl_xor(v, 4);
  v += __shfl_xor(v, 8);
  v += __shfl_xor(v, 16);
  return v;  // full sum broadcast to all 32 lanes
}

__global__ void zero_ws_kernel(unsigned int* __restrict__ w, int n) {
  int i = blockIdx.x * blockDim.x + threadIdx.x;
  if (i < n) w[i] = 0u;
}

// One block = one 256-row chunk of one subset. 256 threads = 8 wave32s.
__global__ void __launch_bounds__(256)
accum_kernel(const float* __restrict__ ps, const float* __restrict__ zs,
             const int* __restrict__ tg,
             float* __restrict__ gSumP, float* __restrict__ gSumZ,
             float* __restrict__ gDiag, int* __restrict__ gCnt) {
  __shared__ float sP[SUMS];   // 64 KB
  __shared__ float sZ[SUMS];   // 64 KB
  __shared__ int   sCnt[NCLS];
  __shared__ float sDiag;

  const int tid = threadIdx.x;
  for (int i = tid; i < SUMS; i += 256) { sP[i] = 0.f; sZ[i] = 0.f; }
  if (tid < NCLS) sCnt[tid] = 0;
  if (tid == 0) sDiag = 0.f;
  __syncthreads();

  const int subset = blockIdx.x >> 4;   // /CHUNKS
  const int chunk  = blockIdx.x & 15;
  const int wave   = tid >> 5;
  const int lane   = tid & 31;

  float dacc = 0.f;

  for (int r = 0; r < 32; ++r) {
    const int row = chunk * ROWS_PER_BLOCK + wave * 32 + r;
    const size_t g = (size_t)subset * MROWS + row;
    const float4* pv = (const float4*)(ps + g * DIMS);
    const float4* zv = (const float4*)(zs + g * DIMS);
    // lane owns 8 consecutive dims: [lane*8, lane*8+7]
    const float4 p0 = pv[lane * 2], p1 = pv[lane * 2 + 1];
    const float4 z0 = zv[lane * 2], z1 = zv[lane * 2 + 1];

    float sp = p0.x*p0.x + p0.y*p0.y + p0.z*p0.z + p0.w*p0.w
             + p1.x*p1.x + p1.y*p1.y + p1.z*p1.z + p1.w*p1.w;
    float sz = z0.x*z0.x + z0.y*z0.y + z0.z*z0.z + z0.w*z0.w
             + z1.x*z1.x + z1.y*z1.y + z1.z*z1.z + z1.w*z1.w;
    float pz = p0.x*z0.x + p0.y*z0.y + p0.z*z0.z + p0.w*z0.w
             + p1.x*z1.x + p1.y*z1.y + p1.z*z1.z + p1.w*z1.w;
    sp = wred32(sp);
    sz = wred32(sz);
    pz = wred32(pz);

    const float invp = 1.f / fmaxf(sqrtf(sp), 1e-8f);
    const float invz = 1.f / fmaxf(sqrtf(sz), 1e-8f);
    dacc += pz * invp * invz;   // identical across lanes

    const int c = tg[g];
    float* dp = &sP[c * DIMS + lane * 8];
    float* dz = &sZ[c * DIMS + lane * 8];
    atomicAdd(&dp[0], p0.x * invp); atomicAdd(&dp[1], p0.y * invp);
    atomicAdd(&dp[2], p0.z * invp); atomicAdd(&dp[3], p0.w * invp);
    atomicAdd(&dp[4], p1.x * invp); atomicAdd(&dp[5], p1.y * invp);
    atomicAdd(&dp[6], p1.z * invp); atomicAdd(&dp[7], p1.w * invp);
    atomicAdd(&dz[0], z0.x * invz); atomicAdd(&dz[1], z0.y * invz);
    atomicAdd(&dz[2], z0.z * invz); atomicAdd(&dz[3], z0.w * invz);
    atomicAdd(&dz[4], z1.x * invz); atomicAdd(&dz[5], z1.y * invz);
    atomicAdd(&dz[6], z1.z * invz); atomicAdd(&dz[7], z1.w * invz);
    if (lane == 0) atomicAdd(&sCnt[c], 1);
  }

  if (lane == 0) atomicAdd(&sDiag, dacc);
  __syncthreads();

  // flush block-local accumulators to global
  float* GP = gSumP + subset * SUMS;
  float* GZ = gSumZ + subset * SUMS;
  for (int i = tid; i < SUMS; i += 256) {
    atomicAdd(&GP[i], sP[i]);
    atomicAdd(&GZ[i], sZ[i]);
  }
  if (tid < NCLS) atomicAdd(&gCnt[subset * NCLS + tid], sCnt[tid]);
  if (tid == 0) atomicAdd(&gDiag[subset], sDiag);
}

// Single wave32. classSum_s = diag of the 64x64 class Gram matrix S_p S_z^T,
// computed with V_WMMA_F32_16X16X4_F32 (exact f32 accumulate).
// A (16x4 f32) layout: lanes 0-15 hold M=lane, K={kb,kb+1}; lanes 16-31 hold
// M=lane-16, K={kb+2,kb+3}. B (4x16) mirrors it, so A/B load identically.
__global__ void __launch_bounds__(32)
finalize_kernel(const float* __restrict__ gSumP, const float* __restrict__ gSumZ,
                const float* __restrict__ gDiag, const int* __restrict__ gCnt,
                float* __restrict__ out) {
  const int lane = threadIdx.x;
  float loss = 0.f;

  for (int s = 0; s < NSUB; ++s) {
    // cnt = sum_c n_c(n_c-1)/2  (lane covers classes lane and lane+32)
    const int n1 = gCnt[s * NCLS + lane];
    const int n2 = gCnt[s * NCLS + 32 + lane];
    float cpart = 0.5f * (float)(n1 * (n1 - 1) + n2 * (n2 - 1));
    const float cnt = wred32(cpart);

    float classSum = 0.f;
    const int lm   = lane & 15;
    const int koff = (lane < 16) ? 0 : 2;

    for (int tile = 0; tile < 4; ++tile) {
      const int cls = tile * 16 + lm;
      const float* prow = gSumP + (size_t)s * SUMS + cls * DIMS + koff;
      const float* zrow = gSumZ + (size_t)s * SUMS + cls * DIMS + koff;
      v8f acc = {};
      for (int kb = 0; kb < DIMS; kb += 4) {
        v2f a, b;
        a.x = prow[kb]; a.y = prow[kb + 1];
        b.x = zrow[kb]; b.y = zrow[kb + 1];
        acc = __builtin_amdgcn_wmma_f32_16x16x4_f32(
            /*neg_a=*/false, a, /*neg_b=*/false, b,
            /*c_mod=*/(short)0, acc, /*reuse_a=*/false, /*reuse_b=*/false);
      }
      // extract Gram diagonal from the 16x16 f32 C/D layout:
      // VGPR r: lanes 0-15 -> M=r, lanes 16-31 -> M=r+8; N = lane&15
      float arr[8];
#pragma unroll
      for (int i = 0; i < 8; ++i) arr[i] = acc[i];
      float d = 0.f;
      if (lane < 16) { if (lm < 8) d = arr[lm]; }
      else           { if (lm >= 8) d = arr[lm - 8]; }
      classSum += wred32(d);
    }

    const float pair = classSum - gDiag[s];
    loss += -0.5f * pair / cnt;
  }

  if (lane == 0) out[0] = loss * (1.0f / (float)NSUB);
}

extern "C" void kernel_launch(void* const* d_in, const int* in_sizes, int n_in,
                              void* d_out, int out_size, void* d_ws, size_t ws_size,
                              hipStream_t stream) {
  const float* ps = (const float*)d_in[0];
  const float* zs = (const float*)d_in[1];
  const int*   tg = (const int*)d_in[2];

  // workspace layout (floats): sumP[3*16384] | sumZ[3*16384] | diag[3] | cnt[3*64] (ints)
  float* F     = (float*)d_ws;
  float* gSumP = F;
  float* gSumZ = F + NSUB * SUMS;
  float* gDiag = F + 2 * NSUB * SUMS;
  int*   gCnt  = (int*)(F + 2 * NSUB * SUMS + NSUB);

  const int words = 2 * NSUB * SUMS + NSUB + NSUB * NCLS;  // 98499
  zero_ws_kernel<<<(words + 255) / 256, 256, 0, stream>>>((unsigned int*)d_ws, words);
  accum_kernel<<<NSUB * CHUNKS, 256, 0, stream>>>(ps, zs, tg, gSumP, gSumZ, gDiag, gCnt);
  finalize_kernel<<<1, 32, 0, stream>>>(gSumP, gSumZ, gDiag, gCnt, (float*)d_out);
}